// MiddleOut_32066225832017
// MI455X (gfx1250) — compile-verified
//
#include <hip/hip_runtime.h>
#include <hip/hip_bf16.h>

// Problem constants (from reference): B=16384, P=32, L=256, R=256, IN=513.
#define PP  32
#define LL  256
#define RR  256
#define IN_ 513

typedef __attribute__((ext_vector_type(2))) float v2f;
typedef __attribute__((ext_vector_type(8))) float v8f;

// ---------------------------------------------------------------------------
// Stage 1: peer reduction (memory-bound, sets the runtime: streams 536 MB).
//   s_b = sum_p m_bp ; q_b = sum_p m_bp^2 ; z_b = sum_p m_bp * peer[b,p,:]
//   x[b, 0:256]   = s_b * my_latent[b,:]   (pairs with Wm = W[:, :256])
//   x[b, 256:512] = z_b                    (pairs with Wp = W[:, 256:512])
// One 256-thread block per b; thread l streams column l across all 32 peers
// (fully coalesced: consecutive lanes hit consecutive addresses per p).
// ---------------------------------------------------------------------------
__global__ void __launch_bounds__(LL)
mo_reduce_peers(const float* __restrict__ my_latent,
                const float* __restrict__ peer_latents,
                const float* __restrict__ peer_metrics,
                float* __restrict__ xbuf,
                float* __restrict__ sbuf,
                float* __restrict__ qbuf) {
  const int b = blockIdx.x;
  const int l = threadIdx.x;                       // 0..255 == L-1
  const float* mrow = peer_metrics + (size_t)b * PP;
  const float* prow = peer_latents + ((size_t)b * PP) * LL + l;

  float s = 0.0f, q = 0.0f, z = 0.0f;
#pragma unroll
  for (int p = 0; p < PP; ++p) {
    const float m = mrow[p];                       // 128B row, L2/L0 resident
    s += m;
    q += m * m;
    z += m * prow[(size_t)p * LL];
  }

  const float u = s * my_latent[(size_t)b * LL + l];
  xbuf[(size_t)b * (2 * LL) + l]      = u;
  xbuf[(size_t)b * (2 * LL) + LL + l] = z;
  if (l == 0) { sbuf[b] = s; qbuf[b] = q; }
}

// ---------------------------------------------------------------------------
// Stage 2: out[b,r] = (1/P) * ( x[b,:] . Wc[r,:] + s_b*bias[r] + q_b*wm[r] )
// where Wc[r,k] = W[r,k] for k<512 and wm[r] = W[r,512].
// V_WMMA_F32_16X16X4_F32 tiles: grid = B/16 blocks of 256 threads (8 waves);
// wave w owns N-tiles at r0 = 32w and 32w+16; K=512 swept 4 at a time.
// A-frag (16x4 f32, ISA layout): lane -> m = lane&15, kb = (lane>>4)*2,
//   {V0,V1} = x[m][kb], x[m][kb+1]  (aligned float2 load, row stride 512).
// B-frag (4x16): lane -> n = lane&15, same kb; {V0,V1} = W[r0+n][kb..kb+1]
//   (row stride 513 is odd -> two scalar b32 loads).
// C/D (16x16 f32): VGPR v -> row v + (lane>=16 ? 8:0), col lane&15.
// ---------------------------------------------------------------------------
__global__ void __launch_bounds__(256)
mo_gemm_wmma(const float* __restrict__ xbuf,
             const float* __restrict__ sbuf,
             const float* __restrict__ qbuf,
             const float* __restrict__ W,
             const float* __restrict__ bias,
             float* __restrict__ out) {
  const int lane = threadIdx.x & 31;
  const int wave = threadIdx.x >> 5;               // 0..7
  const int mt   = blockIdx.x * 16;                // M-tile base row
  const int n    = lane & 15;
  const int kb   = (lane >> 4) * 2;                // 0 or 2

  const float* xrow = xbuf + (size_t)(mt + (lane & 15)) * (2 * LL);
  const int    r0   = wave * 32;
  const float* w0   = W + (size_t)(r0 + n) * IN_;
  const float* w1   = W + (size_t)(r0 + 16 + n) * IN_;

  v8f acc0 = {};
  v8f acc1 = {};

  for (int kk = 0; kk < 2 * LL; kk += 4) {
    const v2f a = *(const v2f*)(xrow + kk + kb);   // 8B aligned: kk+kb even
    v2f b0, b1;
    b0.x = w0[kk + kb];  b0.y = w0[kk + kb + 1];
    b1.x = w1[kk + kb];  b1.y = w1[kk + kb + 1];
    // 8 args: (neg_a, A, neg_b, B, c_mod, C, reuse_a, reuse_b)
    acc0 = __builtin_amdgcn_wmma_f32_16x16x4_f32(false, a, false, b0,
                                                 (short)0, acc0, false, false);
    acc1 = __builtin_amdgcn_wmma_f32_16x16x4_f32(false, a, false, b1,
                                                 (short)0, acc1, false, false);
  }

  const float invP = 1.0f / (float)PP;
  const int   moff = (lane >> 4) * 8;              // C/D row offset per half
#pragma unroll
  for (int v = 0; v < 8; ++v) {
    const int   row = mt + moff + v;
    const float s   = sbuf[row];
    const float q   = qbuf[row];
    const float wmr0 = W[(size_t)(r0 + n) * IN_ + 2 * LL];
    const float wmr1 = W[(size_t)(r0 + 16 + n) * IN_ + 2 * LL];
    out[(size_t)row * RR + r0 + n] =
        invP * (acc0[v] + s * bias[r0 + n] + q * wmr0);
    out[(size_t)row * RR + r0 + 16 + n] =
        invP * (acc1[v] + s * bias[r0 + 16 + n] + q * wmr1);
  }
}

extern "C" void kernel_launch(void* const* d_in, const int* in_sizes, int n_in,
                              void* d_out, int out_size, void* d_ws, size_t ws_size,
                              hipStream_t stream) {
  const float* my_latent    = (const float*)d_in[0];   // (B, L)
  const float* peer_latents = (const float*)d_in[1];   // (B, P, L)
  const float* peer_metrics = (const float*)d_in[2];   // (B, P)
  const float* W            = (const float*)d_in[3];   // (R, 513)
  const float* bias         = (const float*)d_in[4];   // (R,)
  float*       out          = (float*)d_out;           // (B, R)

  const int B = in_sizes[0] / LL;                      // 16384

  // Workspace layout: x (B x 512) | s (B) | q (B)  -> ~33.7 MB of f32.
  float* xbuf = (float*)d_ws;
  float* sbuf = xbuf + (size_t)B * (2 * LL);
  float* qbuf = sbuf + B;

  mo_reduce_peers<<<B, LL, 0, stream>>>(my_latent, peer_latents, peer_metrics,
                                        xbuf, sbuf, qbuf);
  mo_gemm_wmma<<<B / 16, 256, 0, stream>>>(xbuf, sbuf, qbuf, W, bias, out);
}